// generator_1949915152710
// MI455X (gfx1250) — compile-verified
//
#include <hip/hip_runtime.h>

#define C_COUNT 32
#define NC 48
#define ROW 512
#define COL 1024

typedef __attribute__((ext_vector_type(2))) float v2f;
typedef __attribute__((ext_vector_type(8))) float v8f;

// ---- f32 WMMA 16x16x4 helpers (full f32 precision; D = A*B + C) -----------

__device__ __forceinline__ v8f wmma4(v2f a, v2f b, v8f c) {
  // 8 args: (neg_a, A, neg_b, B, c_mod, C, reuse_a, reuse_b)
  return __builtin_amdgcn_wmma_f32_16x16x4_f32(
      /*neg_a=*/false, a, /*neg_b=*/false, b,
      /*c_mod=*/(short)0, c, /*reuse_a=*/false, /*reuse_b=*/false);
}

// A-operand tile (16x4, MxK) from row-major matrix, leading dim ld.
// Lane L<16 : { M[rowBase+L][kBase+0], M[rowBase+L][kBase+1] }
// Lane L>=16: { M[rowBase+L-16][kBase+2], M[rowBase+L-16][kBase+3] }
__device__ __forceinline__ v2f load_a(const float* __restrict__ p, int ld,
                                      int rowBase, int kBase, int lane) {
  const float* q = p + (size_t)(rowBase + (lane & 15)) * ld
                     + kBase + ((lane >> 4) << 1);
  v2f r; r.x = q[0]; r.y = q[1];
  return r;
}

// B-operand tile (4x16, KxN) from row-major KxN matrix, leading dim ld.
// Lane L<16 : { M[kBase+0][nBase+L], M[kBase+1][nBase+L] }
// Lane L>=16: { M[kBase+2][nBase+L-16], M[kBase+3][nBase+L-16] }
__device__ __forceinline__ v2f load_b(const float* __restrict__ p, int ld,
                                      int kBase, int nBase, int lane) {
  const float* q = p + (size_t)(kBase + ((lane >> 4) << 1)) * ld
                     + nBase + (lane & 15);
  v2f r; r.x = q[0]; r.y = q[ld];
  return r;
}

// ---------------------------------------------------------------------------
// K1: d[c,i] = diag(Pb @ Ev @ W_pie)[i].
//     One WG per (c, 16-row block). 8 waves; wave w computes the 16x64 panel
//     U[:, w*64 : w*64+64] of U = Pb[rows]@Ev with WMMA, then contracts it
//     against W_pie columns, accumulating into LDS with ds_add_f32.
// ---------------------------------------------------------------------------
__global__ __launch_bounds__(256) void k_diag(const float* __restrict__ Pb,
                                              const float* __restrict__ Ev,
                                              const float* __restrict__ Wp,
                                              float* __restrict__ d_vec) {
  const int c    = blockIdx.x >> 5;
  const int i0   = (blockIdx.x & 31) << 4;
  const int lane = threadIdx.x & 31;
  const int w    = threadIdx.x >> 5;

  const float* pb = Pb + (size_t)c * ROW * ROW;
  const float* ev = Ev + (size_t)c * ROW * ROW;
  const float* wp = Wp + (size_t)c * ROW * ROW;

  __shared__ float dsum[16];
  if (threadIdx.x < 16) dsum[threadIdx.x] = 0.0f;
  __syncthreads();

  v8f acc[4] = { (v8f)0.0f, (v8f)0.0f, (v8f)0.0f, (v8f)0.0f };
  const int nb = w << 6;  // this wave's 64-column slab of U

  for (int kk = 0; kk < ROW; kk += 4) {
    const v2f a = load_a(pb, ROW, i0, kk, lane);
#pragma unroll
    for (int t = 0; t < 4; ++t) {
      const v2f b = load_b(ev, ROW, kk, nb + (t << 4), lane);
      acc[t] = wmma4(a, b, acc[t]);
    }
  }

  // Contract U[:, k] with W_pie[k, i0+m] and reduce into dsum[m].
  const int hi  = lane >> 4;
  const int nlo = lane & 15;
#pragma unroll
  for (int t = 0; t < 4; ++t) {
    const int kg = nb + (t << 4) + nlo;  // global middle index of this column
#pragma unroll
    for (int v = 0; v < 8; ++v) {
      const int m = v + (hi << 3);
      atomicAdd(&dsum[m], acc[t][v] * wp[(size_t)kg * ROW + i0 + m]);
    }
  }
  __syncthreads();
  if (threadIdx.x < 16) d_vec[c * ROW + i0 + threadIdx.x] = dsum[threadIdx.x];
}

// ---------------------------------------------------------------------------
// K2: Fn = normalize(center((1 + 0.3*d_row) * feat_row)). One wave per row;
//     32 floats per lane held in registers, wave32 __shfl_xor reductions.
// ---------------------------------------------------------------------------
__global__ __launch_bounds__(256) void k_norm(const float* __restrict__ feat,
                                              const float* __restrict__ d_vec,
                                              float* __restrict__ Fn) {
  const int r    = blockIdx.x * 8 + (threadIdx.x >> 5);
  const int lane = threadIdx.x & 31;
  const float scale = 1.0f + 0.3f * d_vec[r];

  const float* src = feat + (size_t)r * COL;
  float*       dst = Fn   + (size_t)r * COL;

  float vals[32];
  float s = 0.0f;
#pragma unroll
  for (int j = 0; j < 8; ++j) {
    const float4 t = *(const float4*)(src + (j << 7) + (lane << 2));
    vals[4 * j + 0] = scale * t.x;
    vals[4 * j + 1] = scale * t.y;
    vals[4 * j + 2] = scale * t.z;
    vals[4 * j + 3] = scale * t.w;
    s += vals[4 * j + 0] + vals[4 * j + 1] + vals[4 * j + 2] + vals[4 * j + 3];
  }
#pragma unroll
  for (int off = 16; off >= 1; off >>= 1) s += __shfl_xor(s, off, 32);
  const float mean = s * (1.0f / COL);

  float q = 0.0f;
#pragma unroll
  for (int e = 0; e < 32; ++e) {
    const float ce = vals[e] - mean;
    vals[e] = ce;
    q += ce * ce;
  }
#pragma unroll
  for (int off = 16; off >= 1; off >>= 1) q += __shfl_xor(q, off, 32);
  const float inv = rsqrtf(q);

#pragma unroll
  for (int j = 0; j < 8; ++j) {
    float4 t;
    t.x = vals[4 * j + 0] * inv;
    t.y = vals[4 * j + 1] * inv;
    t.z = vals[4 * j + 2] * inv;
    t.w = vals[4 * j + 3] * inv;
    *(float4*)(dst + (j << 7) + (lane << 2)) = t;
  }
}

// ---------------------------------------------------------------------------
// K3: generator_W[c] = Fn[c] @ Fn[c]^T.  Since B = A^T of the same matrix,
//     the B-operand tile is loaded with the A-layout from Fn rows nBase..+15.
//     WG = 256 threads = 8 waves covering a 128x64 output tile; each wave
//     owns 32x32 (2x2 accumulator tiles): 4 loads + 4 WMMAs per K-step.
// ---------------------------------------------------------------------------
__global__ __launch_bounds__(256) void k_gram(const float* __restrict__ Fn,
                                              float* __restrict__ out) {
  const int c     = blockIdx.x;
  const int m0    = blockIdx.y << 7;  // 128-row tile
  const int n0    = blockIdx.z << 6;  // 64-col tile
  const int lane  = threadIdx.x & 31;
  const int w     = threadIdx.x >> 5;
  const int mBase = m0 + ((w >> 1) << 5);
  const int nBase = n0 + ((w & 1) << 5);

  const float* f = Fn + (size_t)c * ROW * COL;

  v8f acc[2][2] = { { (v8f)0.0f, (v8f)0.0f }, { (v8f)0.0f, (v8f)0.0f } };

  for (int kk = 0; kk < COL; kk += 4) {
    const v2f a0 = load_a(f, COL, mBase,      kk, lane);
    const v2f a1 = load_a(f, COL, mBase + 16, kk, lane);
    const v2f b0 = load_a(f, COL, nBase,      kk, lane);  // B = Fn^T tile
    const v2f b1 = load_a(f, COL, nBase + 16, kk, lane);
    acc[0][0] = wmma4(a0, b0, acc[0][0]);
    acc[0][1] = wmma4(a0, b1, acc[0][1]);
    acc[1][0] = wmma4(a1, b0, acc[1][0]);
    acc[1][1] = wmma4(a1, b1, acc[1][1]);
  }

  float* o = out + (size_t)c * ROW * ROW;
  const int hi  = lane >> 4;
  const int nlo = lane & 15;
#pragma unroll
  for (int ti = 0; ti < 2; ++ti)
#pragma unroll
    for (int tj = 0; tj < 2; ++tj)
#pragma unroll
      for (int v = 0; v < 8; ++v) {
        const int m = mBase + (ti << 4) + v + (hi << 3);
        const int n = nBase + (tj << 4) + nlo;
        o[(size_t)m * ROW + n] = acc[ti][tj][v];
      }
}

// ---------------------------------------------------------------------------

extern "C" void kernel_launch(void* const* d_in, const int* in_sizes, int n_in,
                              void* d_out, int out_size, void* d_ws, size_t ws_size,
                              hipStream_t stream) {
  // Inputs: 0=Pa 1=Ec 2=Wc (unused) 3=Pb 4=Ev 5=W_pie 6=feat
  const float* Pb   = (const float*)d_in[3];
  const float* Ev   = (const float*)d_in[4];
  const float* Wp   = (const float*)d_in[5];
  const float* feat = (const float*)d_in[6];
  float* out = (float*)d_out;

  // Workspace layout: [0, 64KB) = d[c,i]; [64KB, ...) = Fn (32*512*1024 f32)
  float* d_vec = (float*)d_ws;
  float* Fn    = (float*)((char*)d_ws + 65536);

  k_diag<<<dim3(C_COUNT * (ROW / 16)), 256, 0, stream>>>(Pb, Ev, Wp, d_vec);
  k_norm<<<dim3(C_COUNT * ROW / 8), 256, 0, stream>>>(feat, d_vec, Fn);
  k_gram<<<dim3(C_COUNT, ROW / 128, ROW / 64), 256, 0, stream>>>(Fn, out);
}